// PMLoss_35141422416213
// MI455X (gfx1250) — compile-verified
//
#include <hip/hip_runtime.h>

#define NB 32
#define NC 22
#define NP 2048

typedef __attribute__((ext_vector_type(2))) float v2f;
typedef __attribute__((ext_vector_type(8))) float v8f;

__device__ __forceinline__ float sl1(float x) {
    float ax = fabsf(x);
    return ax < 1.0f ? 0.5f * x * x : ax - 0.5f;
}

__global__ void pm_zero(float* out) {
    if (threadIdx.x == 0 && blockIdx.x == 0) out[0] = 0.0f;
}

// One block per sample: pick class, build rotation matrices, rotate points.
// pred4 = (-2x, -2y, -2z, 1)  ;  gt4 = (x, y, z, |q|^2)
// so that WMMA(pred4_rows, gt4_cols) = |q|^2 - 2 p.q  (argmin objective).
__global__ __launch_bounds__(256) void pm_prep(
    const float* __restrict__ pred, const float* __restrict__ tgt,
    const float* __restrict__ wgt, const float* __restrict__ pts,
    const float* __restrict__ sym,
    float4* __restrict__ pred4, float4* __restrict__ gt4,
    int* __restrict__ flags, float* __restrict__ scale)
{
    const int b = blockIdx.x;

    int cls = 0; bool found = false;
    for (int c = 0; c < NC; ++c) {
        if (wgt[b * 4 * NC + c * 4] > 0.0f) { cls = c; found = true; break; }
    }

    const float* qp = pred + b * 4 * NC + cls * 4;
    const float* qg = tgt  + b * 4 * NC + cls * 4;
    float pw = qp[0], px = qp[1], py = qp[2], pz = qp[3];
    float gw = qg[0], gx = qg[1], gy = qg[2], gz = qg[3];

    float Rp[9], Rg[9];
    Rp[0] = 1.f - 2.f * (py * py + pz * pz); Rp[1] = 2.f * (px * py - pz * pw); Rp[2] = 2.f * (px * pz + py * pw);
    Rp[3] = 2.f * (px * py + pz * pw); Rp[4] = 1.f - 2.f * (px * px + pz * pz); Rp[5] = 2.f * (py * pz - px * pw);
    Rp[6] = 2.f * (px * pz - py * pw); Rp[7] = 2.f * (py * pz + px * pw); Rp[8] = 1.f - 2.f * (px * px + py * py);

    Rg[0] = 1.f - 2.f * (gy * gy + gz * gz); Rg[1] = 2.f * (gx * gy - gz * gw); Rg[2] = 2.f * (gx * gz + gy * gw);
    Rg[3] = 2.f * (gx * gy + gz * gw); Rg[4] = 1.f - 2.f * (gx * gx + gz * gz); Rg[5] = 2.f * (gy * gz - gx * gw);
    Rg[6] = 2.f * (gx * gz - gy * gw); Rg[7] = 2.f * (gy * gz + gx * gw); Rg[8] = 1.f - 2.f * (gx * gx + gy * gy);

    float qdot = fabsf(pw * gw + px * gx + py * gy + pz * gz);
    qdot = fminf(fmaxf(qdot, 0.0f), 1.0f);
    float ang = 2.0f * acosf(qdot) * 57.29577951308232f;   // degrees
    bool usec = (sym[cls] > 0.0f) && (ang > 15.0f);

    if (threadIdx.x == 0) {
        flags[b] = usec ? 1 : 0;
        scale[b] = found ? (1.0f / (float)(NB * NP)) : 0.0f;
    }

    const float* base = pts + (size_t)cls * NP * 3;
    for (int p = threadIdx.x; p < NP; p += blockDim.x) {
        float x = base[p * 3 + 0], y = base[p * 3 + 1], z = base[p * 3 + 2];
        float ax = Rp[0] * x + Rp[1] * y + Rp[2] * z;
        float ay = Rp[3] * x + Rp[4] * y + Rp[5] * z;
        float az = Rp[6] * x + Rp[7] * y + Rp[8] * z;
        pred4[(size_t)b * NP + p] = make_float4(-2.f * ax, -2.f * ay, -2.f * az, 1.0f);
        float bx = Rg[0] * x + Rg[1] * y + Rg[2] * z;
        float by = Rg[3] * x + Rg[4] * y + Rg[5] * z;
        float bz = Rg[6] * x + Rg[7] * y + Rg[8] * z;
        gt4[(size_t)b * NP + p] = make_float4(bx, by, bz, bx * bx + by * by + bz * bz);
    }
}

// grid = (NP/128, NB); 256 threads = 8 waves; wave w handles row tile m0.
// Each wave: 128 x v_wmma_f32_16x16x4_f32 over the LDS-staged gt points,
// running (min, argmin) per row, shfl-xor reduction, smooth-L1, atomic add.
__global__ __launch_bounds__(256) void pm_nn(
    const float4* __restrict__ pred4, const float4* __restrict__ gt4,
    const int* __restrict__ flags, const float* __restrict__ scale,
    float* __restrict__ out)
{
    __shared__ float4 sgt[NP];               // 32 KB of this sample's gt points

    const int b = blockIdx.y;
    const float4* gtb = gt4 + (size_t)b * NP;
    for (int i = threadIdx.x; i < NP; i += blockDim.x) sgt[i] = gtb[i];
    __syncthreads();

    const int wave = threadIdx.x >> 5;
    const int lane = threadIdx.x & 31;
    const int idx  = lane & 15;              // column / row-in-tile index
    const int half = lane >> 4;              // 0: K0,K1 ; 1: K2,K3
    const int m0   = (blockIdx.x * 8 + wave) * 16;

    // A tile: 16 rows of (-2x,-2y,-2z,1); lanes 0-15 carry (K0,K1), 16-31 (K2,K3)
    float4 av = pred4[(size_t)b * NP + m0 + idx];
    v2f a;
    a.x = half ? av.z : av.x;
    a.y = half ? av.w : av.y;

    float best[8];
    int   bi[8];
#pragma unroll
    for (int r = 0; r < 8; ++r) { best[r] = 3.0e38f; bi[r] = 0; }

    for (int n0 = 0; n0 < NP; n0 += 16) {
        float4 bv = sgt[n0 + idx];
        v2f bb;
        bb.x = half ? bv.z : bv.x;
        bb.y = half ? bv.w : bv.y;
        v8f c = {};
        v8f d = __builtin_amdgcn_wmma_f32_16x16x4_f32(
            false, a, false, bb, (short)0, c, false, false);
#pragma unroll
        for (int r = 0; r < 8; ++r) {
            bool lt = d[r] < best[r];
            best[r] = lt ? d[r] : best[r];
            bi[r]   = lt ? (n0 + idx) : bi[r];
        }
    }

    // min-reduce across the 16 lanes of each half (masks 8..1 stay in-half)
#pragma unroll
    for (int off = 8; off >= 1; off >>= 1) {
#pragma unroll
        for (int r = 0; r < 8; ++r) {
            float ob = __shfl_xor(best[r], off, 32);
            int   oi = __shfl_xor(bi[r],   off, 32);
            bool lt = (ob < best[r]) || (ob == best[r] && oi < bi[r]);
            best[r] = lt ? ob : best[r];
            bi[r]   = lt ? oi : bi[r];
        }
    }

    if (idx == 0) {                          // lanes 0 and 16 own the row results
        const bool usec = flags[b] != 0;
        float sum = 0.0f;
#pragma unroll
        for (int r = 0; r < 8; ++r) {
            int M = m0 + r + half * 8;       // C/D layout: half 1 holds rows +8
            float4 pv = pred4[(size_t)b * NP + M];
            float ppx = -0.5f * pv.x, ppy = -0.5f * pv.y, ppz = -0.5f * pv.z;
            float4 gv = usec ? sgt[bi[r]] : sgt[M];
            sum += sl1(ppx - gv.x) + sl1(ppy - gv.y) + sl1(ppz - gv.z);
        }
        atomicAdd(out, sum * scale[b]);
    }
}

extern "C" void kernel_launch(void* const* d_in, const int* in_sizes, int n_in,
                              void* d_out, int out_size, void* d_ws, size_t ws_size,
                              hipStream_t stream) {
    (void)in_sizes; (void)n_in; (void)out_size; (void)ws_size;
    const float* pred = (const float*)d_in[0];
    const float* tgt  = (const float*)d_in[1];
    const float* wgt  = (const float*)d_in[2];
    const float* pts  = (const float*)d_in[3];
    const float* sym  = (const float*)d_in[4];
    float* out = (float*)d_out;

    float4* pred4 = (float4*)d_ws;                       // 1 MB
    float4* gt4   = pred4 + (size_t)NB * NP;             // 1 MB
    int*    flags = (int*)(gt4 + (size_t)NB * NP);
    float*  scale = (float*)(flags + NB);

    pm_zero<<<1, 32, 0, stream>>>(out);
    pm_prep<<<NB, 256, 0, stream>>>(pred, tgt, wgt, pts, sym,
                                    pred4, gt4, flags, scale);
    dim3 grid(NP / (16 * 8), NB);                        // (16, 32)
    pm_nn<<<grid, 256, 0, stream>>>(pred4, gt4, flags, scale, out);
}